// GaussianImage_RS_19224273617586
// MI455X (gfx1250) — compile-verified
//
#include <hip/hip_runtime.h>
#include <math.h>

#define HH 128
#define WW 128
#define NPTS 2048
#define DIMU 768
#define PIX (HH * WW)          // 16384
#define GSTRIDE 12             // floats per gaussian record (48 B)

typedef __attribute__((ext_vector_type(2))) float v2f;
typedef __attribute__((ext_vector_type(8))) float v8f;

// ---------------------------------------------------------------------------
// Stage 1: per-gaussian activations -> conic + center + opacity + color (AoS)
// record: [A, B, C, gx, gy, op, fr, fg, fb, 0, 0, 0]
// ---------------------------------------------------------------------------
__global__ void precompute_kernel(const float* __restrict__ xyz,
                                  const float* __restrict__ scaling,
                                  const float* __restrict__ rotation,
                                  const float* __restrict__ features,
                                  const float* __restrict__ opacity,
                                  float* __restrict__ gs) {
  int n = blockIdx.x * blockDim.x + threadIdx.x;
  if (n >= NPTS) return;
  float mx = tanhf(xyz[2 * n]);
  float my = tanhf(xyz[2 * n + 1]);
  float s0 = fabsf(scaling[2 * n] + 0.5f);
  float s1 = fabsf(scaling[2 * n + 1] + 0.5f);
  float theta = (1.0f / (1.0f + __expf(-rotation[n]))) * 6.283185307179586f;
  float sn = __sinf(theta);
  float c  = __cosf(theta);
  float a = c * s0, b = -sn * s1, d = sn * s0, e = c * s1;
  float cxx = a * a + b * b;
  float cxy = a * d + b * e;
  float cyy = d * d + e * e;
  float inv = 1.0f / (cxx * cyy - cxy * cxy);
  float* g = gs + n * GSTRIDE;
  g[0] = cyy * inv;            // A
  g[1] = -cxy * inv;           // B
  g[2] = cxx * inv;            // C
  g[3] = 0.5f * (mx + 1.0f) * (float)WW;  // gx
  g[4] = 0.5f * (my + 1.0f) * (float)HH;  // gy
  g[5] = opacity[n];
  g[6] = features[3 * n + 0];
  g[7] = features[3 * n + 1];
  g[8] = features[3 * n + 2];
  g[9] = 0.0f; g[10] = 0.0f; g[11] = 0.0f;
}

// ---------------------------------------------------------------------------
// Stage 2: summation rasterizer. One thread per pixel; the gaussian record is
// read with a wave-uniform index so the compiler can scalarize to s_load
// broadcasts. Output is channel-major imgC[4][PIX] with channel 3 == 0 so it
// feeds the WMMA B-matrix (K padded 3 -> 4) directly.
// ---------------------------------------------------------------------------
__global__ void __launch_bounds__(256) raster_kernel(const float* __restrict__ gs,
                                                     float* __restrict__ imgC) {
  int p = blockIdx.x * blockDim.x + threadIdx.x;
  float px = (float)(p & (WW - 1)) + 0.5f;
  float py = (float)(p >> 7) + 0.5f;
  float accr = 0.0f, accg = 0.0f, accb = 0.0f;
  for (int n = 0; n < NPTS; ++n) {
    const float* gp = gs + n * GSTRIDE;
    float A = gp[0], B = gp[1], C = gp[2];
    float gx = gp[3], gy = gp[4], op = gp[5];
    float fr = gp[6], fg = gp[7], fb = gp[8];
    float dx = gx - px;
    float dy = gy - py;
    float sigma = 0.5f * (A * dx * dx + C * dy * dy) + B * dx * dy;
    float al = fminf(0.999f, op * __expf(-sigma));
    bool ok = (sigma >= 0.0f) & (al >= (1.0f / 255.0f));
    al = ok ? al : 0.0f;
    accr += al * fr;
    accg += al * fg;
    accb += al * fb;
  }
  imgC[0 * PIX + p] = fminf(fmaxf(accr, 0.0f), 1.0f);
  imgC[1 * PIX + p] = fminf(fmaxf(accg, 0.0f), 1.0f);
  imgC[2 * PIX + p] = fminf(fmaxf(accb, 0.0f), 1.0f);
  imgC[3 * PIX + p] = 0.0f;   // zero K=3 row for the WMMA pad
}

// ---------------------------------------------------------------------------
// Stage 3: out[DIM, PIX] = w_up[DIM,3] @ imgC[3,PIX] + b_up, via
// V_WMMA_F32_16X16X4_F32.  A = 16x4 w_up tile (M=dim), B = 4x16 imgC tile
// (N=pixel), C = bias.  Each wave: one dim-tile x four pixel-tiles (A and
// bias reused; 4 static v_wmma per wave).  Grid covers the tile space
// exactly, so EXEC is all ones at every WMMA.
// ---------------------------------------------------------------------------
__global__ void __launch_bounds__(256) upproj_wmma_kernel(const float* __restrict__ w_up,
                                                          const float* __restrict__ b_up,
                                                          const float* __restrict__ imgC,
                                                          float* __restrict__ out) {
  const int lane = threadIdx.x & 31;
  const int wave = (int)((blockIdx.x * blockDim.x + threadIdx.x) >> 5);
  const int pixGroups = PIX / 64;            // 256 groups of 4 pixel-tiles
  const int dimTile = wave / pixGroups;      // 0..47
  const int pixGroup = wave % pixGroups;     // 0..255
  const int dimBase = dimTile * 16;
  const int pixBase = pixGroup * 64;
  const int half = lane >> 4;                // 0: lanes 0-15, 1: lanes 16-31
  const int l16 = lane & 15;
  const int k0 = half ? 2 : 0;               // K pair held by this half-wave

  // A matrix 16x4 (M=dim, K=channel): VGPR0 holds K={0|2}, VGPR1 K={1|3}.
  const int dimA = dimBase + l16;
  v2f a;
  a.x = w_up[dimA * 3 + k0];
  a.y = half ? 0.0f : w_up[dimA * 3 + 1];    // K=3 column is zero pad

  // C/D layout: VGPR r -> dim = dimBase + r + 8*half, lane l16 -> pixel.
  const int dimC = dimBase + half * 8;
  v8f c;
#pragma unroll
  for (int r = 0; r < 8; ++r) c[r] = b_up[dimC + r];

#pragma unroll
  for (int t = 0; t < 4; ++t) {
    const int pix = pixBase + t * 16 + l16;
    // B matrix 4x16 (K=channel, N=pixel): VGPR0 K={0|2}, VGPR1 K={1|3};
    // imgC row 3 is stored as zeros, so the uniform indexing is safe.
    v2f b;
    b.x = imgC[(k0 + 0) * PIX + pix];
    b.y = imgC[(k0 + 1) * PIX + pix];
    v8f d = __builtin_amdgcn_wmma_f32_16x16x4_f32(
        /*neg_a=*/false, a, /*neg_b=*/false, b,
        /*c_mod=*/(short)0, c, /*reuse_a=*/false, /*reuse_b=*/false);
#pragma unroll
    for (int r = 0; r < 8; ++r) {
      out[(size_t)(dimC + r) * PIX + pix] = d[r];
    }
  }
}

// ---------------------------------------------------------------------------
extern "C" void kernel_launch(void* const* d_in, const int* in_sizes, int n_in,
                              void* d_out, int out_size, void* d_ws, size_t ws_size,
                              hipStream_t stream) {
  // inputs: 0:x (unused) 1:xyz 2:scaling 3:rotation 4:features 5:opacity 6:w_up 7:b_up
  const float* xyz      = (const float*)d_in[1];
  const float* scaling  = (const float*)d_in[2];
  const float* rotation = (const float*)d_in[3];
  const float* features = (const float*)d_in[4];
  const float* opacity  = (const float*)d_in[5];
  const float* w_up     = (const float*)d_in[6];
  const float* b_up     = (const float*)d_in[7];
  float* out = (float*)d_out;

  float* gs   = (float*)d_ws;                 // 2048 * 12 floats = 96 KB
  float* imgC = gs + NPTS * GSTRIDE;          // 4 * 16384 floats = 256 KB

  precompute_kernel<<<(NPTS + 255) / 256, 256, 0, stream>>>(
      xyz, scaling, rotation, features, opacity, gs);

  raster_kernel<<<PIX / 256, 256, 0, stream>>>(gs, imgC);

  // 48 dim-tiles * 256 pixel-groups = 12288 waves = 1536 blocks of 256
  const int waves = (DIMU / 16) * (PIX / 64);
  upproj_wmma_kernel<<<(waves * 32) / 256, 256, 0, stream>>>(
      w_up, b_up, imgC, out);
}